// Model_80092550135832
// MI455X (gfx1250) — compile-verified
//
#include <hip/hip_runtime.h>

typedef __attribute__((ext_vector_type(2))) float v2f;
typedef __attribute__((ext_vector_type(8))) float v8f;

#define D    64
#define KC   128     // k-chunk rows of lin_W staged in LDS
#define STI  (KC+4)  // input-GEMM LDS col stride (dwords): 132 % 64 == 4 -> conflict-free b64
#define STL  (D+4)   // layer-GEMM LDS col stride (dwords):  68 % 64 == 4 -> conflict-free b64
#define TMI  2       // M-tiles per wave, input GEMM
#define TML  4       // M-tiles per wave, layer GEMM

// ================= CSR build (replaces 600M feature atomics with 3.2M index atomics) ========
__global__ void count_edges_int(const int* __restrict__ dst, int* __restrict__ deg, int nE) {
    int e = blockIdx.x * blockDim.x + threadIdx.x;
    if (e < nE) atomicAdd(&deg[dst[e]], 1);
}

__global__ void scan_block(const int* __restrict__ deg, int* __restrict__ ex,
                           int* __restrict__ bsum, int n) {
    __shared__ int sh[256];
    int tid = threadIdx.x;
    int i = blockIdx.x * 256 + tid;
    int v = (i < n) ? deg[i] : 0;
    sh[tid] = v;
    __syncthreads();
    for (int off = 1; off < 256; off <<= 1) {
        int add = (tid >= off) ? sh[tid - off] : 0;
        __syncthreads();
        sh[tid] += add;
        __syncthreads();
    }
    if (i < n) ex[i] = sh[tid] - v;          // exclusive scan within block
    if (tid == 255) bsum[blockIdx.x] = sh[255];
}

__global__ void scan_sums(int* __restrict__ bsum, int nb) {   // nb <= 1024 (N <= 262144)
    __shared__ int sh[1024];
    int tid = threadIdx.x;
    int v = (tid < nb) ? bsum[tid] : 0;
    sh[tid] = v;
    __syncthreads();
    for (int off = 1; off < 1024; off <<= 1) {
        int add = (tid >= off) ? sh[tid - off] : 0;
        __syncthreads();
        sh[tid] += add;
        __syncthreads();
    }
    if (tid < nb) bsum[tid] = sh[tid] - v;   // exclusive
}

__global__ void add_offsets(int* __restrict__ ex, const int* __restrict__ bsum,
                            int* __restrict__ cursor, int n) {
    int i = blockIdx.x * 256 + threadIdx.x;
    if (i < n) {
        int r = ex[i] + bsum[i >> 8];
        ex[i] = r;        // row start
        cursor[i] = r;    // fill cursor; becomes row end after fill_csr
    }
}

__global__ void fill_csr(const int* __restrict__ src, const int* __restrict__ dst,
                         int* __restrict__ cursor, int* __restrict__ srcs, int nE) {
    int e = blockIdx.x * blockDim.x + threadIdx.x;
    if (e < nE) {
        int pos = atomicAdd(&cursor[dst[e]], 1);
        srcs[pos] = src[e];
    }
}

// ================= atomic-free mean aggregation (16 lanes per destination node) =============
__global__ void aggregate_csr(const float* __restrict__ x, const int* __restrict__ srcs,
                              const int* __restrict__ rowStart, const int* __restrict__ rowEnd,
                              float* __restrict__ agg, int n) {
    int t = blockIdx.x * blockDim.x + threadIdx.x;
    int node = t >> 4;
    if (node >= n) return;
    int c = (t & 15) * 4;
    int s0 = rowStart[node], s1 = rowEnd[node];
    float4 acc = make_float4(0.f, 0.f, 0.f, 0.f);
    for (int i = s0; i < s1; ++i) {
        int s = srcs[i];
        float4 v = *(const float4*)(x + (size_t)s * D + c);
        acc.x += v.x; acc.y += v.y; acc.z += v.z; acc.w += v.w;
    }
    float inv = 1.0f / fmaxf((float)(s1 - s0), 1.0f);
    acc.x *= inv; acc.y *= inv; acc.z *= inv; acc.w *= inv;
    *(float4*)(agg + (size_t)node * D + c) = acc;
}

// ================= input gathers ================
__global__ void gather_rows(const float* __restrict__ emb, const int* __restrict__ nid,
                            float* __restrict__ out, int n16) {
    int t = blockIdx.x * blockDim.x + threadIdx.x;
    if (t >= n16) return;
    int i = t >> 4, c = (t & 15) * 4;
    int s = nid[i];
    *(float4*)(out + (size_t)i * D + c) = *(const float4*)(emb + (size_t)s * D + c);
}

// ====== input GEMM: xg = go_x @ lin_W + lin_b + go_emb[nid]  (fp32 WMMA, LDS B^T) ==========
__global__ __launch_bounds__(256) void input_gemm_wmma(
        const float* __restrict__ go_x, const float* __restrict__ lin_W,
        const float* __restrict__ lin_b, const float* __restrict__ go_emb,
        const int* __restrict__ go_nid, float* __restrict__ xg, int M, int K) {
    __shared__ float sWt[D * STI];           // transposed chunk: sWt[col*STI + k], 33 KB

    int waveInBlk = threadIdx.x >> 5;
    int lane = threadIdx.x & 31;
    int row  = lane & 15;
    int kh   = lane >> 4;
    int colL = lane & 15;

    int numTiles = M >> 4;
    int tile0 = (blockIdx.x * 8 + waveInBlk) * TMI;

    int   m0c[TMI];
    const float* arow[TMI];
    bool  valid[TMI];
#pragma unroll
    for (int t = 0; t < TMI; ++t) {
        int ti = tile0 + t;
        valid[t] = (ti < numTiles);
        m0c[t] = valid[t] ? (ti << 4) : 0;               // clamp: loads stay legal, EXEC stays full
        arow[t] = go_x + (size_t)(m0c[t] + row) * K;
    }

    v8f acc[TMI][4];
#pragma unroll
    for (int t = 0; t < TMI; ++t)
#pragma unroll
        for (int j = 0; j < 4; ++j) {
            float b = lin_b[j * 16 + colL];
#pragma unroll
            for (int g = 0; g < 8; ++g) acc[t][j][g] = b;
        }

    for (int kc0 = 0; kc0 < K; kc0 += KC) {
        int len = (K - kc0 < KC) ? (K - kc0) : KC;       // F_GO: len always multiple of 4
        __syncthreads();
        for (int i = threadIdx.x * 4; i < len * D; i += 256 * 4) {
            int r = i >> 6, c = i & 63;                  // transpose while staging
            float4 w = *(const float4*)&lin_W[(size_t)(kc0 + r) * D + c];
            sWt[(c + 0) * STI + r] = w.x;
            sWt[(c + 1) * STI + r] = w.y;
            sWt[(c + 2) * STI + r] = w.z;
            sWt[(c + 3) * STI + r] = w.w;
        }
        __syncthreads();
        if (kc0 + KC < K) {                              // prefetch next A chunk (speculative)
#pragma unroll
            for (int t = 0; t < TMI; ++t)
                __builtin_prefetch(arow[t] + kc0 + KC, 0, 0);
        }
        for (int k = 0; k < len; k += 4) {
            v2f a[TMI];
#pragma unroll
            for (int t = 0; t < TMI; ++t)
                a[t] = *(const v2f*)(arow[t] + kc0 + k + 2 * kh);
#pragma unroll
            for (int j = 0; j < 4; ++j) {
                // one ds_load_b64: k-adjacent pair, bank-conflict-free (STI % 64 == 4)
                v2f b = *(const v2f*)&sWt[(j * 16 + colL) * STI + k + 2 * kh];
#pragma unroll
                for (int t = 0; t < TMI; ++t)
                    acc[t][j] = __builtin_amdgcn_wmma_f32_16x16x4_f32(
                        false, a[t], false, b, (short)0, acc[t][j], false, false);
            }
        }
    }

    int hi = kh * 8;
#pragma unroll
    for (int t = 0; t < TMI; ++t) {
        if (!valid[t]) continue;                         // wave-uniform
#pragma unroll
        for (int g = 0; g < 8; ++g) {
            int m = m0c[t] + g + hi;
            int nid = go_nid[m];
#pragma unroll
            for (int j = 0; j < 4; ++j) {
                int col = j * 16 + colL;
                xg[(size_t)m * D + col] = acc[t][j][g] + go_emb[(size_t)nid * D + col];
            }
        }
    }
}

// ====== layer GEMM: out = agg @ Wl + bl + x @ Wr, optional ReLU (fp32 WMMA, LDS W^T) =======
__global__ __launch_bounds__(256) void layer_gemm_wmma(
        const float* __restrict__ agg, const float* __restrict__ x,
        const float* __restrict__ Wl, const float* __restrict__ bl,
        const float* __restrict__ Wr, float* __restrict__ out, int M, int relu) {
    __shared__ float sWt[2 * D * STL];       // Wl^T then Wr^T, 34 KB total

    for (int i = threadIdx.x * 4; i < D * D; i += 256 * 4) {
        int r = i >> 6, c = i & 63;          // transpose while staging
        float4 wl = *(const float4*)&Wl[i];
        float4 wr = *(const float4*)&Wr[i];
        sWt[(c + 0) * STL + r] = wl.x;
        sWt[(c + 1) * STL + r] = wl.y;
        sWt[(c + 2) * STL + r] = wl.z;
        sWt[(c + 3) * STL + r] = wl.w;
        sWt[D * STL + (c + 0) * STL + r] = wr.x;
        sWt[D * STL + (c + 1) * STL + r] = wr.y;
        sWt[D * STL + (c + 2) * STL + r] = wr.z;
        sWt[D * STL + (c + 3) * STL + r] = wr.w;
    }
    __syncthreads();

    int waveInBlk = threadIdx.x >> 5;
    int lane = threadIdx.x & 31;
    int row  = lane & 15;
    int kh   = lane >> 4;
    int colL = lane & 15;

    int numTiles = M >> 4;
    int tile0 = (blockIdx.x * 8 + waveInBlk) * TML;

    int   m0c[TML];
    bool  valid[TML];
    const float* aggrow[TML];
    const float* xrow[TML];
#pragma unroll
    for (int t = 0; t < TML; ++t) {
        int ti = tile0 + t;
        valid[t] = (ti < numTiles);
        m0c[t] = valid[t] ? (ti << 4) : 0;
        aggrow[t] = agg + (size_t)(m0c[t] + row) * D;
        xrow[t]   = x   + (size_t)(m0c[t] + row) * D;
    }

    v8f acc[TML][4];
#pragma unroll
    for (int t = 0; t < TML; ++t)
#pragma unroll
        for (int j = 0; j < 4; ++j) {
            float b = bl[j * 16 + colL];
#pragma unroll
            for (int g = 0; g < 8; ++g) acc[t][j][g] = b;
        }

    for (int k = 0; k < D; k += 4) {
        v2f aA[TML], aX[TML];
#pragma unroll
        for (int t = 0; t < TML; ++t) {
            aA[t] = *(const v2f*)(aggrow[t] + k + 2 * kh);
            aX[t] = *(const v2f*)(xrow[t]   + k + 2 * kh);
        }
#pragma unroll
        for (int j = 0; j < 4; ++j) {
            v2f bL = *(const v2f*)&sWt[(j * 16 + colL) * STL + k + 2 * kh];
            v2f bR = *(const v2f*)&sWt[D * STL + (j * 16 + colL) * STL + k + 2 * kh];
#pragma unroll
            for (int t = 0; t < TML; ++t) {
                acc[t][j] = __builtin_amdgcn_wmma_f32_16x16x4_f32(
                    false, aA[t], false, bL, (short)0, acc[t][j], false, false);
                acc[t][j] = __builtin_amdgcn_wmma_f32_16x16x4_f32(
                    false, aX[t], false, bR, (short)0, acc[t][j], false, false);
            }
        }
    }

    int hi = kh * 8;
#pragma unroll
    for (int t = 0; t < TML; ++t) {
        if (!valid[t]) continue;                         // wave-uniform
#pragma unroll
        for (int g = 0; g < 8; ++g) {
            int m = m0c[t] + g + hi;
#pragma unroll
            for (int j = 0; j < 4; ++j) {
                float v = acc[t][j][g];
                if (relu) v = fmaxf(v, 0.0f);
                out[(size_t)m * D + j * 16 + colL] = v;
            }
        }
    }
}

// ================= classifier ================
__global__ void classifier_dot(const float* __restrict__ xp, const float* __restrict__ xg,
                               const int* __restrict__ ls, const int* __restrict__ ld,
                               float* __restrict__ out, int n) {
    int e = blockIdx.x * blockDim.x + threadIdx.x;
    if (e >= n) return;
    const float4* a = (const float4*)(xp + (size_t)ls[e] * D);
    const float4* b = (const float4*)(xg + (size_t)ld[e] * D);
    float s = 0.0f;
#pragma unroll
    for (int i = 0; i < 16; ++i) {
        float4 va = a[i], vb = b[i];
        s += va.x * vb.x + va.y * vb.y + va.z * vb.z + va.w * vb.w;
    }
    out[e] = s;
}

// ================= host orchestration ================
extern "C" void kernel_launch(void* const* d_in, const int* in_sizes, int n_in,
                              void* d_out, int out_size, void* d_ws, size_t ws_size,
                              hipStream_t stream) {
    const float* go_x  = (const float*)d_in[0];
    const float* p_emb = (const float*)d_in[1];
    const float* g_emb = (const float*)d_in[2];
    const float* lin_W = (const float*)d_in[3];
    const float* lin_b = (const float*)d_in[4];
    const float* Wl    = (const float*)d_in[5];
    const float* bl    = (const float*)d_in[6];
    const float* Wr    = (const float*)d_in[7];
    const int* p_nid   = (const int*)d_in[8];
    const int* g_nid   = (const int*)d_in[9];
    const int* src_pg  = (const int*)d_in[10];
    const int* dst_pg  = (const int*)d_in[11];
    const int* src_gp  = (const int*)d_in[12];
    const int* dst_gp  = (const int*)d_in[13];
    const int* lab_s   = (const int*)d_in[14];
    const int* lab_d   = (const int*)d_in[15];

    const int NP  = in_sizes[8];
    const int NG  = in_sizes[9];
    const int E   = in_sizes[10];
    const int EL  = in_sizes[14];
    const int FGO = in_sizes[0] / NG;

    // ---- workspace carve-up (~122 MB) ----
    char* base = (char*)d_ws;
    auto alloc = [&](size_t bytes) -> char* {
        char* p = base;
        base += (bytes + 255) & ~(size_t)255;
        return p;
    };
    float* xp_a  = (float*)alloc((size_t)NP * D * 4);
    float* xp_b  = (float*)alloc((size_t)NP * D * 4);
    float* xg_a  = (float*)alloc((size_t)NG * D * 4);
    float* xg_b  = (float*)alloc((size_t)NG * D * 4);
    float* agg_p = (float*)alloc((size_t)NP * D * 4);
    float* agg_g = (float*)alloc((size_t)NG * D * 4);
    int* row_g   = (int*)alloc((size_t)NG * 4);
    int* cur_g   = (int*)alloc((size_t)NG * 4);
    int* row_p   = (int*)alloc((size_t)NP * 4);
    int* cur_p   = (int*)alloc((size_t)NP * 4);
    int* srcs_g  = (int*)alloc((size_t)E * 4);
    int* srcs_p  = (int*)alloc((size_t)E * 4);
    int* bsum    = (int*)alloc(1024 * 4);

    const int EB = (E + 255) / 256;

    // ---- CSR build, direction protein->go (rows = GO nodes) ----
    {
        int nb = (NG + 255) / 256;
        hipMemsetAsync(cur_g, 0, (size_t)NG * 4, stream);
        count_edges_int<<<EB, 256, 0, stream>>>(dst_pg, cur_g, E);
        scan_block<<<nb, 256, 0, stream>>>(cur_g, row_g, bsum, NG);
        scan_sums<<<1, 1024, 0, stream>>>(bsum, nb);
        add_offsets<<<nb, 256, 0, stream>>>(row_g, bsum, cur_g, NG);
        fill_csr<<<EB, 256, 0, stream>>>(src_pg, dst_pg, cur_g, srcs_g, E);
    }
    // ---- CSR build, direction go->protein (rows = protein nodes) ----
    {
        int nb = (NP + 255) / 256;
        hipMemsetAsync(cur_p, 0, (size_t)NP * 4, stream);
        count_edges_int<<<EB, 256, 0, stream>>>(dst_gp, cur_p, E);
        scan_block<<<nb, 256, 0, stream>>>(cur_p, row_p, bsum, NP);
        scan_sums<<<1, 1024, 0, stream>>>(bsum, nb);
        add_offsets<<<nb, 256, 0, stream>>>(row_p, bsum, cur_p, NP);
        fill_csr<<<EB, 256, 0, stream>>>(src_gp, dst_gp, cur_p, srcs_p, E);
    }

    // ---- input embeddings ----
    gather_rows<<<(NP * 16 + 255) / 256, 256, 0, stream>>>(p_emb, p_nid, xp_a, NP * 16);
    {
        int tiles  = (NG + 15) / 16;
        int blocks = (tiles + 8 * TMI - 1) / (8 * TMI);
        input_gemm_wmma<<<blocks, 256, 0, stream>>>(go_x, lin_W, lin_b, g_emb, g_nid,
                                                    xg_a, NG, FGO);
    }

    float* xp_cur = xp_a; float* xp_nxt = xp_b;
    float* xg_cur = xg_a; float* xg_nxt = xg_b;
    for (int l = 0; l < 3; ++l) {
        aggregate_csr<<<(NG * 16 + 255) / 256, 256, 0, stream>>>(xp_cur, srcs_g, row_g, cur_g,
                                                                 agg_g, NG);
        aggregate_csr<<<(NP * 16 + 255) / 256, 256, 0, stream>>>(xg_cur, srcs_p, row_p, cur_p,
                                                                 agg_p, NP);
        const float* Wl0 = Wl + (size_t)(l * 2 + 0) * D * D;
        const float* Wl1 = Wl + (size_t)(l * 2 + 1) * D * D;
        const float* Wr0 = Wr + (size_t)(l * 2 + 0) * D * D;
        const float* Wr1 = Wr + (size_t)(l * 2 + 1) * D * D;
        const float* bl0 = bl + (size_t)(l * 2 + 0) * D;
        const float* bl1 = bl + (size_t)(l * 2 + 1) * D;
        int relu = (l < 2) ? 1 : 0;
        {
            int tiles  = (NG + 15) / 16;
            int blocks = (tiles + 8 * TML - 1) / (8 * TML);
            layer_gemm_wmma<<<blocks, 256, 0, stream>>>(agg_g, xg_cur, Wl0, bl0, Wr0,
                                                        xg_nxt, NG, relu);
        }
        {
            int tiles  = (NP + 15) / 16;
            int blocks = (tiles + 8 * TML - 1) / (8 * TML);
            layer_gemm_wmma<<<blocks, 256, 0, stream>>>(agg_p, xp_cur, Wl1, bl1, Wr1,
                                                        xp_nxt, NP, relu);
        }
        float* t;
        t = xp_cur; xp_cur = xp_nxt; xp_nxt = t;
        t = xg_cur; xg_cur = xg_nxt; xg_nxt = t;
    }

    classifier_dot<<<(EL + 255) / 256, 256, 0, stream>>>(xp_cur, xg_cur, lab_s, lab_d,
                                                         (float*)d_out, EL);
}